// PretrainEncoder_74388833566984
// MI455X (gfx1250) — compile-verified
//
#include <hip/hip_runtime.h>

// GNN encoder for MI455X (gfx1250, wave32).
// Key algebra: (x[src] @ Wm) == (x @ Wm)[src] -> 64x64 GEMMs run at node
// granularity on V_WMMA_F32_16X16X4_F32 (fp32 kept: problem is memory-bound,
// bf16 buys nothing). Edge loop = fused tiny-GEMM + gather + fp32 atomic
// scatter; whole working set (~51MB state + streamed edges) is L2-resident
// (192MB), atomics resolve at L2. We tile staged into LDS via the CDNA5
// async-to-LDS path (ASYNCcnt).

typedef float v2f __attribute__((ext_vector_type(2)));
typedef float v8f __attribute__((ext_vector_type(8)));

#define HID 64
#define EMB 48
#define EDIM 16
#define NLAYER 4

__device__ __forceinline__ v8f wmma4(v2f a, v2f b, v8f c) {
  // D = A(16x4,f32) * B(4x16,f32) + C(16x16,f32)
  return __builtin_amdgcn_wmma_f32_16x16x4_f32(
      /*neg_a=*/false, a, /*neg_b=*/false, b,
      /*c_mod=*/(short)0, c, /*reuse_a=*/false, /*reuse_b=*/false);
}

// ---------------------------------------------------------------------------
// x = (embed_weight[node_idx] * embed_mask[node_idx]) @ W_in   (N x 48 -> N x 64)
// One wave per 16-node tile. Embedding table (96KB) is L0/L2 resident.
// ---------------------------------------------------------------------------
__global__ __launch_bounds__(128)
void embed_gemm(const int* __restrict__ node_idx,
                const float* __restrict__ ew,
                const float* __restrict__ em,
                const float* __restrict__ Win,
                float* __restrict__ x, int N) {
  const int lane = threadIdx.x & 31;
  const int wave = threadIdx.x >> 5;
  const int tile = blockIdx.x * (blockDim.x >> 5) + wave;
  const int row0 = tile * 16;
  if (row0 >= N) return;                 // wave-uniform: EXEC stays all-1s
  const int half = lane >> 4;
  const int mn   = lane & 15;
  const int arow = row0 + mn;            // A-matrix row for this lane
  const bool av  = arow < N;
  const int nidx = av ? node_idx[arow] : 0;
  const float msk = av ? 1.0f : 0.0f;

  v8f acc[4] = {};
  const int ks = 2 * half;
#pragma unroll
  for (int k0 = 0; k0 < EMB; k0 += 4) {
    const int k = k0 + ks;
    v2f a;
    a.x = msk * ew[nidx * EMB + k]     * em[nidx * EMB + k];
    a.y = msk * ew[nidx * EMB + k + 1] * em[nidx * EMB + k + 1];
#pragma unroll
    for (int j = 0; j < 4; ++j) {
      v2f b = { Win[k * HID + j * 16 + mn], Win[(k + 1) * HID + j * 16 + mn] };
      acc[j] = wmma4(a, b, acc[j]);
    }
  }
#pragma unroll
  for (int j = 0; j < 4; ++j)
#pragma unroll
    for (int r = 0; r < 8; ++r) {
      const int row = row0 + r + 8 * half;
      if (row < N) x[row * HID + j * 16 + mn] = acc[j][r];
    }
}

// ---------------------------------------------------------------------------
// P = x @ Wm[l],  S = x @ Wself[l]  — dual GEMM sharing the A fragments.
// ---------------------------------------------------------------------------
__global__ __launch_bounds__(128)
void dual_node_gemm(const float* __restrict__ x,
                    const float* __restrict__ Wm,
                    const float* __restrict__ Ws,
                    float* __restrict__ P,
                    float* __restrict__ S, int N) {
  const int lane = threadIdx.x & 31;
  const int wave = threadIdx.x >> 5;
  const int tile = blockIdx.x * (blockDim.x >> 5) + wave;
  const int row0 = tile * 16;
  if (row0 >= N) return;
  const int half = lane >> 4;
  const int mn   = lane & 15;
  const int arow = row0 + mn;
  const bool av  = arow < N;

  v8f accP[4] = {};
  v8f accS[4] = {};
  const int ks = 2 * half;
#pragma unroll
  for (int k0 = 0; k0 < HID; k0 += 4) {
    const int k = k0 + ks;
    v2f a;
    a.x = av ? x[arow * HID + k]     : 0.0f;
    a.y = av ? x[arow * HID + k + 1] : 0.0f;
#pragma unroll
    for (int j = 0; j < 4; ++j) {
      v2f bm = { Wm[k * HID + j * 16 + mn], Wm[(k + 1) * HID + j * 16 + mn] };
      v2f bs = { Ws[k * HID + j * 16 + mn], Ws[(k + 1) * HID + j * 16 + mn] };
      accP[j] = wmma4(a, bm, accP[j]);
      accS[j] = wmma4(a, bs, accS[j]);
    }
  }
#pragma unroll
  for (int j = 0; j < 4; ++j)
#pragma unroll
    for (int r = 0; r < 8; ++r) {
      const int row = row0 + r + 8 * half;
      if (row < N) {
        P[row * HID + j * 16 + mn] = accP[j][r];
        S[row * HID + j * 16 + mn] = accS[j][r];
      }
    }
}

__global__ void zero_f32v4(float4* __restrict__ p, int n4) {
  int i = blockIdx.x * blockDim.x + threadIdx.x;
  const int stride = gridDim.x * blockDim.x;
  const float4 z = make_float4(0.f, 0.f, 0.f, 0.f);
  for (; i < n4; i += stride) p[i] = z;
}

// ---------------------------------------------------------------------------
// Edge kernel: per edge e, q = edge_attr[e] @ We (16->64), msg = q * P[src],
// agg[dst] += msg via global_atomic_add_f32 (resolved in L2).
// 16 edges per 256-thread block, 16 lanes per edge, 4 cols (float4) per lane:
// b128 gathers, ds_load_b128 weight reads (bank-conflict-free + same-address
// broadcast across half-waves), v_pk_fma accumulation. We tile (4KB) staged
// with global_load_async_to_lds_b128 (ASYNCcnt), no VGPR round-trip.
// ---------------------------------------------------------------------------
__global__ __launch_bounds__(256)
void edge_msg(const int* __restrict__ eidx,     // (2, E): [0..E)=src, [E..2E)=dst
              const float* __restrict__ eattr,  // (E, 16)
              const float* __restrict__ We,     // (16, 64) this layer
              const float* __restrict__ P,
              float* __restrict__ agg, int E) {
  __shared__ float sWe[EDIM * HID];   // 4KB, row-major 16x64
  __shared__ float sAttr[16][EDIM];   // 1KB
  {
    // 256 lanes x 16B = exactly 4KB: one async b128 per lane.
    const float* g = We + threadIdx.x * 4;
    const unsigned lds = (unsigned)(uintptr_t)&sWe[threadIdx.x * 4];
    asm volatile("global_load_async_to_lds_b128 %0, %1, off"
                 :: "v"(lds), "v"(g) : "memory");
  }
  {
    const int ee = blockIdx.x * 16 + (threadIdx.x >> 4);
    sAttr[threadIdx.x >> 4][threadIdx.x & 15] =
        (ee < E) ? eattr[(size_t)ee * EDIM + (threadIdx.x & 15)] : 0.0f;
  }
  asm volatile("s_wait_asynccnt 0x0" ::: "memory");
  __syncthreads();

  const int le = threadIdx.x >> 4;    // edge slot in block (0..15)
  const int cq = threadIdx.x & 15;    // column quad: cols 4*cq .. 4*cq+3
  const int e  = blockIdx.x * 16 + le;
  if (e >= E) return;

  const float4* sWe4 = (const float4*)sWe;
  float4 q = make_float4(0.f, 0.f, 0.f, 0.f);
#pragma unroll
  for (int k = 0; k < EDIM; ++k) {
    const float  a = sAttr[le][k];          // same-bank broadcast
    const float4 w = sWe4[k * 16 + cq];     // ds_load_b128
    q.x += a * w.x; q.y += a * w.y; q.z += a * w.z; q.w += a * w.w;
  }
  const int s = eidx[e];
  const int d = eidx[E + e];
  const float4 p = *(const float4*)(P + (size_t)s * HID + 4 * cq);  // b128 gather
  float* ap = agg + (size_t)d * HID + 4 * cq;
  atomicAdd(ap + 0, q.x * p.x);
  atomicAdd(ap + 1, q.y * p.y);
  atomicAdd(ap + 2, q.z * p.z);
  atomicAdd(ap + 3, q.w * p.w);
}

// ---------------------------------------------------------------------------
// h = agg @ Wn + S (S preloaded into the WMMA accumulator);
// gates = sigmoid(h[:, :16]); x = h * repeat(gates, 4).
// Gate for column c lives in C-fragment 0 at lane half*16 + (c>>2), same VGPR
// row -> fetched with a wave32 shuffle, no LDS round-trip.
// ---------------------------------------------------------------------------
__global__ __launch_bounds__(128)
void h_gemm_gate(const float* __restrict__ agg,
                 const float* __restrict__ Wn,
                 const float* __restrict__ S,
                 float* __restrict__ x, int N) {
  const int lane = threadIdx.x & 31;
  const int wave = threadIdx.x >> 5;
  const int tile = blockIdx.x * (blockDim.x >> 5) + wave;
  const int row0 = tile * 16;
  if (row0 >= N) return;
  const int half = lane >> 4;
  const int mn   = lane & 15;
  const int arow = row0 + mn;
  const bool av  = arow < N;

  v8f acc[4];
#pragma unroll
  for (int j = 0; j < 4; ++j)
#pragma unroll
    for (int r = 0; r < 8; ++r) {
      const int row = row0 + r + 8 * half;
      acc[j][r] = (row < N) ? S[row * HID + j * 16 + mn] : 0.0f;
    }

  const int ks = 2 * half;
#pragma unroll
  for (int k0 = 0; k0 < HID; k0 += 4) {
    const int k = k0 + ks;
    v2f a;
    a.x = av ? agg[arow * HID + k]     : 0.0f;
    a.y = av ? agg[arow * HID + k + 1] : 0.0f;
#pragma unroll
    for (int j = 0; j < 4; ++j) {
      v2f b = { Wn[k * HID + j * 16 + mn], Wn[(k + 1) * HID + j * 16 + mn] };
      acc[j] = wmma4(a, b, acc[j]);
    }
  }

#pragma unroll
  for (int r = 0; r < 8; ++r) {
    const float h0 = acc[0][r];        // fragment 0 holds cols 0..15 (the gates)
    float vals[4];
#pragma unroll
    for (int j = 0; j < 4; ++j) {
      const int gsrc = half * 16 + 4 * j + (mn >> 2);  // lane holding h[row, c>>2]
      const float g  = __shfl(h0, gsrc, 32);
      const float sg = 1.0f / (1.0f + __expf(-g));
      vals[j] = ((j == 0) ? h0 : acc[j][r]) * sg;
    }
    const int row = row0 + r + 8 * half;
    if (row < N) {
#pragma unroll
      for (int j = 0; j < 4; ++j) x[row * HID + j * 16 + mn] = vals[j];
    }
  }
}

__global__ __launch_bounds__(256)
void head_k(const float* __restrict__ x, const float* __restrict__ Wh,
            const float* __restrict__ bh, float* __restrict__ out, int N) {
  const int n = blockIdx.x * blockDim.x + threadIdx.x;
  if (n >= N) return;
  const float4* xr = reinterpret_cast<const float4*>(x + (size_t)n * HID);
  float acc = 0.0f;
#pragma unroll
  for (int i = 0; i < 16; ++i) {
    const float4 v = xr[i];
    acc += v.x * Wh[4 * i] + v.y * Wh[4 * i + 1] + v.z * Wh[4 * i + 2] + v.w * Wh[4 * i + 3];
  }
  out[n] = acc + bh[0];
}

extern "C" void kernel_launch(void* const* d_in, const int* in_sizes, int n_in,
                              void* d_out, int out_size, void* d_ws, size_t ws_size,
                              hipStream_t stream) {
  const int*   node_idx = (const int*)  d_in[0];
  const int*   eidx     = (const int*)  d_in[1];   // (2, E)
  const float* eattr    = (const float*)d_in[2];   // (E, 16)
  // d_in[3] = z, d_in[4] = canonical: unused by the reference math
  const float* ew    = (const float*)d_in[5];      // (512, 48)
  const float* em    = (const float*)d_in[6];      // (512, 48)
  const float* Win   = (const float*)d_in[7];      // (48, 64)
  const float* Wm    = (const float*)d_in[8];      // (4, 64, 64)
  const float* We    = (const float*)d_in[9];      // (4, 16, 64)
  const float* Wself = (const float*)d_in[10];     // (4, 64, 64)
  const float* Wn    = (const float*)d_in[11];     // (4, 64, 64)
  const float* Wh    = (const float*)d_in[12];     // (64, 1)
  const float* bh    = (const float*)d_in[13];     // (1,)
  float* out = (float*)d_out;

  const int N = in_sizes[0];
  const int E = in_sizes[1] / 2;

  float* x   = (float*)d_ws;
  float* P   = x + (size_t)N * HID;
  float* S   = P + (size_t)N * HID;
  float* agg = S + (size_t)N * HID;

  const int WAVES = 4;                       // 128 threads = 4 wave32
  const int tiles = (N + 15) / 16;
  dim3 blkG(32 * WAVES);
  dim3 grdG((tiles + WAVES - 1) / WAVES);

  embed_gemm<<<grdG, blkG, 0, stream>>>(node_idx, ew, em, Win, x, N);

  const int n4 = (N * HID) / 4;
  for (int l = 0; l < NLAYER; ++l) {
    dual_node_gemm<<<grdG, blkG, 0, stream>>>(x, Wm + (size_t)l * HID * HID,
                                              Wself + (size_t)l * HID * HID, P, S, N);
    zero_f32v4<<<(n4 + 255) / 256, 256, 0, stream>>>((float4*)agg, n4);
    edge_msg<<<(E + 15) / 16, 256, 0, stream>>>(eidx, eattr,
                                                We + (size_t)l * EDIM * HID, P, agg, E);
    h_gemm_gate<<<grdG, blkG, 0, stream>>>(agg, Wn + (size_t)l * HID * HID, S, x, N);
  }

  head_k<<<(N + 255) / 256, 256, 0, stream>>>(x, Wh, bh, out, N);
}